// SimpleHGN_final_18580028522785
// MI455X (gfx1250) — compile-verified
//
#include <hip/hip_runtime.h>
#include <hip/hip_bf16.h>
#include <math.h>

typedef __attribute__((ext_vector_type(2))) float v2f;
typedef __attribute__((ext_vector_type(8))) float v8f;

#define NEG_SLOPE 0.2f
#define IN_MAX 512   // reference: IN = 512

// sign-split float atomic max (works with init = -inf)
__device__ __forceinline__ void atomicMaxF32(float* addr, float v) {
    if (v >= 0.f) {
        atomicMax((int*)addr, __float_as_int(v));
    } else {
        atomicMin((unsigned int*)addr, __float_as_uint(v));
    }
}

// ---------------------------------------------------------------------------
// Tiny precompute: wal[k] = (W @ a_l)[k]   (IN)
//                  h_e[t] = edge_emb[t] . (W_e @ a_e)   (NET<=64)
// ---------------------------------------------------------------------------
__global__ void precompute_kernel(const float* __restrict__ W,
                                  const float* __restrict__ W_e,
                                  const float* __restrict__ a_l,
                                  const float* __restrict__ a_e,
                                  const float* __restrict__ edge_emb,
                                  float* __restrict__ wal,
                                  float* __restrict__ h_e,
                                  int IN, int NET) {
    __shared__ float wea[64];
    int tid = threadIdx.x;
    for (int k = tid; k < IN; k += blockDim.x) {
        float s = 0.f;
        #pragma unroll
        for (int c = 0; c < 16; ++c) s += W[k * 16 + c] * a_l[c];
        wal[k] = s;
    }
    if (tid < 64) {
        float s = 0.f;
        #pragma unroll
        for (int j = 0; j < 64; ++j) s += W_e[tid * 64 + j] * a_e[j];
        wea[tid] = s;
    }
    __syncthreads();
    if (tid < NET) {
        float s = 0.f;
        #pragma unroll
        for (int k = 0; k < 64; ++k) s += edge_emb[tid * 64 + k] * wea[k];
        h_e[tid] = s;
    }
}

// ---------------------------------------------------------------------------
// Node GEMMs via WMMA f32 16x16x4: one wave per 16-node tile.
//   c_t = tailF_tile @ W        -> h_tail[N,16], hr = c_t . a_r (via LDS)
//   c_r = headF_tile @ res_W    -> out_acc init (+ res_b)
//   c_a = headF_tile @ [wal|0…] -> hl (column 0), wal masked per-lane (no
//                                  divergence in the hot loop)
// Also initializes mmax = -inf, denom = 0.
// ---------------------------------------------------------------------------
__global__ void __launch_bounds__(32)
node_gemm_kernel(const float* __restrict__ headF,
                 const float* __restrict__ tailF,
                 const float* __restrict__ W,
                 const float* __restrict__ resW,
                 const float* __restrict__ res_b,
                 const float* __restrict__ a_r,
                 const float* __restrict__ wal,
                 float* __restrict__ h_tail,
                 float* __restrict__ hl,
                 float* __restrict__ hr,
                 float* __restrict__ mmax,
                 float* __restrict__ denom,
                 float* __restrict__ out_acc,
                 int N, int IN) {
    __shared__ float s_wal[IN_MAX];
    __shared__ float tile[16][17];

    const int lane  = threadIdx.x;        // 0..31
    const int base  = blockIdx.x * 16;
    const int n     = lane & 15;          // col (B/C) or row (A)
    const int khalf = (lane >> 4) << 1;   // 0 or 2 : K sub-pair per A/B layout
    const float lmask = (n == 0) ? 1.f : 0.f;

    for (int k = lane; k < IN; k += 32) s_wal[k] = wal[k];
    __syncthreads();

    int arow = base + n;
    if (arow >= N) arow = N - 1;          // clamp reads, guard writes
    const float* tailRow = tailF + (size_t)arow * IN;
    const float* headRow = headF + (size_t)arow * IN;

    v8f c_t = {};  // tail @ W
    v8f c_r = {};  // head @ res_W
    v8f c_a = {};  // head @ [wal | 0...]

    for (int k0 = 0; k0 < IN; k0 += 4) {
        const int k = k0 + khalf;
        // A frags: 16x4 f32 (lane = M, half-wave selects K pair) — b64 loads
        const float2 t2 = *(const float2*)(tailRow + k);
        const float2 h2 = *(const float2*)(headRow + k);
        v2f at, ah, bw, br, bl;
        at.x = t2.x; at.y = t2.y;
        ah.x = h2.x; ah.y = h2.y;
        // B frags: 4x16 f32 (lane = N, half-wave selects K pair)
        bw.x = W[k * 16 + n];        bw.y = W[(k + 1) * 16 + n];
        br.x = resW[k * 16 + n];     br.y = resW[(k + 1) * 16 + n];
        bl.x = s_wal[k] * lmask;     bl.y = s_wal[k + 1] * lmask;

        c_t = __builtin_amdgcn_wmma_f32_16x16x4_f32(false, at, false, bw,
                                                    (short)0, c_t, false, false);
        c_r = __builtin_amdgcn_wmma_f32_16x16x4_f32(false, ah, false, br,
                                                    (short)0, c_r, false, false);
        c_a = __builtin_amdgcn_wmma_f32_16x16x4_f32(false, ah, false, bl,
                                                    (short)0, c_a, false, false);
    }

    // C/D layout: VGPR v, lanes 0-15 -> row v, lanes 16-31 -> row v+8, col = n
    const int rhalf = (lane >> 4) * 8;
    #pragma unroll
    for (int v = 0; v < 8; ++v) {
        const int row = base + rhalf + v;
        const float vt = c_t[v];
        tile[rhalf + v][n] = vt;
        if (row < N) {
            h_tail[(size_t)row * 16 + n]  = vt;
            out_acc[(size_t)row * 16 + n] = c_r[v] + res_b[n];
        }
    }
    if (n == 0) {   // column 0 of c_a holds hl
        #pragma unroll
        for (int v = 0; v < 8; ++v) {
            const int row = base + rhalf + v;
            if (row < N) hl[row] = c_a[v];
        }
    }
    __syncthreads();
    if (lane < 16) {
        const int row = base + lane;
        if (row < N) {
            float s = 0.f;
            #pragma unroll
            for (int c = 0; c < 16; ++c) s += a_r[c] * tile[lane][c];
            hr[row]    = s;
            mmax[row]  = -__builtin_inff();
            denom[row] = 0.f;
        }
    }
}

// ---------------------------------------------------------------------------
// Pass 3: per-edge logits + segment max
// ---------------------------------------------------------------------------
__global__ void edge_logit_max_kernel(const int* __restrict__ eidx,
                                      const int* __restrict__ etype,
                                      const float* __restrict__ hl,
                                      const float* __restrict__ hr,
                                      const float* __restrict__ h_e,
                                      float* __restrict__ elog,
                                      float* __restrict__ mmax,
                                      int E) {
    int i = blockIdx.x * blockDim.x + threadIdx.x;
    if (i >= E) return;
    const int h  = eidx[i];
    const int t  = eidx[E + i];
    const int ty = etype[i];
    const float x  = hl[h] + hr[t] + h_e[ty];
    const float lg = x > 0.f ? x : NEG_SLOPE * x;
    elog[i] = lg;
    atomicMaxF32(&mmax[h], lg);
}

// ---------------------------------------------------------------------------
// Pass 4: exp + segment sum
// ---------------------------------------------------------------------------
__global__ void edge_exp_sum_kernel(const int* __restrict__ eidx,
                                    float* __restrict__ elog,   // in logit, out ex
                                    const float* __restrict__ mmax,
                                    float* __restrict__ denom,
                                    int E) {
    int i = blockIdx.x * blockDim.x + threadIdx.x;
    if (i >= E) return;
    const int h = eidx[i];
    const float ex = __expf(elog[i] - mmax[h]);
    elog[i] = ex;
    atomicAdd(&denom[h], ex);
}

// ---------------------------------------------------------------------------
// Pass 5: attn-weighted SpMM aggregation (L2-resident atomics)
// ---------------------------------------------------------------------------
__global__ void edge_aggregate_kernel(const int* __restrict__ eidx,
                                      const float* __restrict__ ex,
                                      const float* __restrict__ denom,
                                      const float* __restrict__ h_tail,
                                      float* __restrict__ out_acc,
                                      int E) {
    int i = blockIdx.x * blockDim.x + threadIdx.x;
    if (i >= E) return;
    const int h = eidx[i];
    const int t = eidx[E + i];
    const float attn = ex[i] / denom[h];
    const float4* src = (const float4*)(h_tail + (size_t)t * 16);
    float* dst = out_acc + (size_t)h * 16;
    #pragma unroll
    for (int q = 0; q < 4; ++q) {
        const float4 v = src[q];
        atomicAdd(dst + q * 4 + 0, attn * v.x);
        atomicAdd(dst + q * 4 + 1, attn * v.y);
        atomicAdd(dst + q * 4 + 2, attn * v.z);
        atomicAdd(dst + q * 4 + 3, attn * v.w);
    }
}

// ---------------------------------------------------------------------------
// Pass 6: L2 row-normalize then softmax (in place in d_out)
// ---------------------------------------------------------------------------
__global__ void finalize_kernel(float* __restrict__ out, int N) {
    int i = blockIdx.x * blockDim.x + threadIdx.x;
    if (i >= N) return;
    float v[16];
    float4* p = (float4*)(out + (size_t)i * 16);
    #pragma unroll
    for (int q = 0; q < 4; ++q) {
        const float4 x = p[q];
        v[q * 4 + 0] = x.x; v[q * 4 + 1] = x.y;
        v[q * 4 + 2] = x.z; v[q * 4 + 3] = x.w;
    }
    float ss = 0.f;
    #pragma unroll
    for (int c = 0; c < 16; ++c) ss += v[c] * v[c];
    const float inv = 1.f / fmaxf(sqrtf(ss), 1e-12f);
    float mx = -__builtin_inff();
    #pragma unroll
    for (int c = 0; c < 16; ++c) { v[c] *= inv; mx = fmaxf(mx, v[c]); }
    float s = 0.f;
    #pragma unroll
    for (int c = 0; c < 16; ++c) { v[c] = __expf(v[c] - mx); s += v[c]; }
    const float invs = 1.f / s;
    #pragma unroll
    for (int q = 0; q < 4; ++q) {
        float4 x;
        x.x = v[q * 4 + 0] * invs; x.y = v[q * 4 + 1] * invs;
        x.z = v[q * 4 + 2] * invs; x.w = v[q * 4 + 3] * invs;
        p[q] = x;
    }
}

// ---------------------------------------------------------------------------
extern "C" void kernel_launch(void* const* d_in, const int* in_sizes, int n_in,
                              void* d_out, int out_size, void* d_ws, size_t ws_size,
                              hipStream_t stream) {
    const float* headF    = (const float*)d_in[0];
    const float* tailF    = (const float*)d_in[1];
    const int*   eidx     = (const int*)d_in[2];
    const int*   etype    = (const int*)d_in[3];
    const float* W        = (const float*)d_in[4];
    const float* W_e      = (const float*)d_in[5];
    const float* a_l      = (const float*)d_in[6];
    const float* a_r      = (const float*)d_in[7];
    const float* a_e      = (const float*)d_in[8];
    const float* edge_emb = (const float*)d_in[9];
    const float* resW     = (const float*)d_in[10];
    const float* res_b    = (const float*)d_in[11];

    const int C   = 16;
    const int IN  = in_sizes[4] / C;        // 512
    const int N   = in_sizes[0] / IN;       // 100000
    const int E   = in_sizes[3];            // 3200000
    const int NET = in_sizes[9] / 64;       // 8

    // workspace layout (floats)
    float* ws     = (float*)d_ws;
    float* wal    = ws;                     // IN
    float* h_e    = wal + IN;               // 64
    float* hl     = h_e + 64;               // N
    float* hr     = hl + N;                 // N
    float* mmax   = hr + N;                 // N
    float* denom  = mmax + N;               // N
    float* h_tail = denom + N;              // N*16
    float* elog   = h_tail + (size_t)N * 16;// E
    float* out    = (float*)d_out;          // N*16 accumulator + final

    precompute_kernel<<<1, 512, 0, stream>>>(W, W_e, a_l, a_e, edge_emb,
                                             wal, h_e, IN, NET);

    node_gemm_kernel<<<(N + 15) / 16, 32, 0, stream>>>(
        headF, tailF, W, resW, res_b, a_r, wal,
        h_tail, hl, hr, mmax, denom, out, N, IN);

    const int eb = 256;
    const int eg = (E + eb - 1) / eb;
    edge_logit_max_kernel<<<eg, eb, 0, stream>>>(eidx, etype, hl, hr, h_e,
                                                 elog, mmax, E);
    edge_exp_sum_kernel<<<eg, eb, 0, stream>>>(eidx, elog, mmax, denom, E);
    edge_aggregate_kernel<<<eg, eb, 0, stream>>>(eidx, elog, denom, h_tail,
                                                 out, E);
    finalize_kernel<<<(N + 255) / 256, 256, 0, stream>>>(out, N);
}